// MultiQCDAware_64948495450697
// MI455X (gfx1250) — compile-verified
//
#include <hip/hip_runtime.h>
#include <math.h>

typedef __attribute__((ext_vector_type(2))) float v2f;
typedef __attribute__((ext_vector_type(4))) float v4f;
typedef __attribute__((ext_vector_type(8))) float v8f;

#define NN 1024
#define Bn 4
#define Fn 8
#define Dn 6

__device__ __forceinline__ unsigned umin32(unsigned a, unsigned b) { return a < b ? a : b; }

// Monotone float<->uint maps so unsigned atomicMax == float max (any sign).
__device__ __forceinline__ unsigned fmono(float f) {
    unsigned u = __float_as_uint(f);
    return (u & 0x80000000u) ? ~u : (u | 0x80000000u);
}
__device__ __forceinline__ float funmono(unsigned u) {
    return __uint_as_float((u & 0x80000000u) ? (u & 0x7FFFFFFFu) : ~u);
}

// ---------------------------------------------------------------------------
// Kernel 1: precompute pm[b][f][n] = momentum^(2*alpha_f); init dmin slots.
// ---------------------------------------------------------------------------
__global__ __launch_bounds__(256) void k_pm_init(
    const float* __restrict__ emb, const float* __restrict__ alpha,
    float* __restrict__ pm, unsigned* __restrict__ dminb) {
    int idx = blockIdx.x * 256 + threadIdx.x;
    if (idx < Bn * Fn) dminb[idx] = 0x7F800000u;  // +inf bits (nonneg float order)
    if (idx < Bn * Fn * NN) {
        int n = idx & (NN - 1);
        int f = (idx >> 10) & (Fn - 1);
        int b = idx >> 13;
        float m = emb[(b * Dn + 4) * NN + n];
        pm[idx] = __expf(2.0f * alpha[f] * __logf(m));
    }
}

// ---------------------------------------------------------------------------
// Kernel 2: dmin[b][f] = min over i!=j of sqdist(i,j)*min(pm_i,pm_j).
// Block = (i-band of 16, b). 256 threads: 16 threads per row.
// All values are nonneg floats -> uint compares == float compares, so the
// whole min-reduction runs on v_min_u32 (no canonicalize ops).
// ---------------------------------------------------------------------------
__global__ __launch_bounds__(256) void k_dmin(
    const float* __restrict__ emb, const float* __restrict__ pm,
    unsigned* __restrict__ dminb) {
    __shared__ float ep[NN][2];    // eta,phi interleaved (8KB)
    __shared__ float pmt[NN][8];   // [j][f] (32KB) -> 2x ds_load_b128 per j
    __shared__ unsigned smin[8];

    const int tid = threadIdx.x;
    const int it = blockIdx.x, b = blockIdx.y;

    for (int idx = tid; idx < NN; idx += 256) {
        ep[idx][0] = emb[(b * Dn + 1) * NN + idx];
        ep[idx][1] = emb[(b * Dn + 2) * NN + idx];
    }
    for (int idx = tid; idx < NN * 8; idx += 256) {
        int ff = idx >> 10, n = idx & (NN - 1);
        pmt[n][ff] = pm[(b * Fn + ff) * NN + n];
    }
    if (tid < 8) smin[tid] = 0x7F800000u;
    __syncthreads();

    const int r = tid >> 4;          // row within band 0..15
    const int sub = tid & 15;        // j-phase
    const int ig = it * 16 + r;
    const float ei = ep[ig][0], pi = ep[ig][1];
    const float PINF = __uint_as_float(0x7F800000u);
    unsigned pmiu[8], lminu[8];
#pragma unroll
    for (int f = 0; f < 8; f++) { pmiu[f] = __float_as_uint(pmt[ig][f]); lminu[f] = 0x7F800000u; }

    for (int j = sub; j < NN; j += 16) {
        float de = ei - ep[j][0];
        float dp = pi - ep[j][1];
        float sq = fmaf(de, de, dp * dp);
        sq = (j == ig) ? PINF : sq;     // exclude diagonal, branchless
#pragma unroll
        for (int f = 0; f < 8; f++) {
            unsigned mmu = umin32(pmiu[f], __float_as_uint(pmt[j][f]));  // v_min_u32
            float d = sq * __uint_as_float(mmu);
            lminu[f] = umin32(lminu[f], __float_as_uint(d));             // v_min_u32
        }
    }
#pragma unroll
    for (int f = 0; f < 8; f++) atomicMin(&smin[f], lminu[f]);
    __syncthreads();
    if (tid < 8) atomicMin(&dminb[b * Fn + tid], smin[tid]);
}

// ---------------------------------------------------------------------------
// Kernel 3: softmax rows. Block = (i-band 16, f, b), 8 waves x 32 lanes.
// WMMA computes D[M=j][N=i] per 16x16 tile:
//   A row j = [eta_j, phi_j, r_j, 1], B col i = [-2 eta_i, -2 phi_i, 1, r_i]
// Lane holds one column i = i0+l15, rows j = j0+mbase+0..7 => per-lane scalar
// softmax stats for one row i and two b128 stores per tile. t-values are kept
// pre-scaled by log2(e) so the exp pass is a bare v_exp_f32.
// ---------------------------------------------------------------------------
__global__ __launch_bounds__(256) void k_softmax(
    const float* __restrict__ emb, const float* __restrict__ beta,
    const float* __restrict__ pm, const unsigned* __restrict__ dminb,
    float* __restrict__ out) {
    __shared__ v2f Apack[2 * NN];   // [n]={eta,phi}, [NN+n]={r,1}       (16KB)
    __shared__ v2f Bpack[2 * NN];   // [n]={-2eta,-2phi}, [NN+n]={1,r}   (16KB)
    __shared__ float pm_s[NN];      // (4KB)
    __shared__ unsigned maxb[16];
    __shared__ float rmax_s[16], rsum_s[16], inv_s[16];

    const int tid = threadIdx.x;
    const int it = blockIdx.x, f = blockIdx.y, b = blockIdx.z;

    for (int idx = tid; idx < NN; idx += 256) {
        float e = emb[(b * Dn + 1) * NN + idx];
        float p = emb[(b * Dn + 2) * NN + idx];
        float r = fmaf(e, e, p * p);
        v2f a0; a0.x = e; a0.y = p;
        v2f a1; a1.x = r; a1.y = 1.0f;
        v2f b0; b0.x = -2.0f * e; b0.y = -2.0f * p;
        v2f b1; b1.x = 1.0f; b1.y = r;
        Apack[idx] = a0; Apack[NN + idx] = a1;
        Bpack[idx] = b0; Bpack[NN + idx] = b1;
        pm_s[idx] = pm[(b * Fn + f) * NN + idx];
    }
    if (tid < 16) { maxb[tid] = 0u; rsum_s[tid] = 0.0f; }

    const float L2E = 1.4426950408889634f;
    const float dmin = __uint_as_float(dminb[b * Fn + f]);
    const float bb = beta[f];
    const float denom = dmin + ((dmin == 0.0f) ? 1e-9f : 0.0f);
    const float c2 = (bb * bb / denom) * L2E;     // exp2-domain slope
    const float tc2 = c2 * dmin;                  // exp2-domain offset
    const float NEGINF = __uint_as_float(0xFF800000u);
    __syncthreads();

    const int lane = tid & 31;
    const int wave = tid >> 5;
    const int half = lane >> 4;
    const int l15 = lane & 15;
    const int i0 = it * 16;
    const int mbase = half * 8;           // D rows (j-local) this lane holds
    const int hoff = half * NN;           // half-select by address, no branches

    const v2f Bv = Bpack[hoff + i0 + l15];        // fixed per wave: column i
    const unsigned pmiu = __float_as_uint(pm_s[i0 + l15]);

    float tv[8][8];                                // [j-tile][k] in VGPRs
    float lmax = NEGINF;

#pragma unroll
    for (int tt = 0; tt < 8; tt++) {
        const int jt = wave * 8 + tt;
        const int j0 = jt * 16;
        const v2f Av = Apack[hoff + j0 + l15];    // per-tile: rows j
        v8f Cz = {};
        v8f Dv = __builtin_amdgcn_wmma_f32_16x16x4_f32(
            false, Av, false, Bv, (short)0, Cz, false, false);
        const v4f pj0 = *(const __attribute__((address_space(3))) v4f*)&pm_s[j0 + mbase];
        const v4f pj1 = *(const __attribute__((address_space(3))) v4f*)&pm_s[j0 + mbase + 4];
#pragma unroll
        for (int k = 0; k < 8; k++) {
            float pmj = (k < 4) ? pj0[k & 3] : pj1[k & 3];
            unsigned mmu = umin32(pmiu, __float_as_uint(pmj));   // v_min_u32 (pm>0)
            float d = Dv[k] * __uint_as_float(mmu);
            tv[tt][k] = tc2 - c2 * d;             // log2-domain logit, single FMA
        }
        if (jt == it) {                            // wave-uniform: diagonal tile
#pragma unroll
            for (int k = 0; k < 8; k++)
                if (l15 == mbase + k) tv[tt][k] = NEGINF;
        }
#pragma unroll
        for (int k = 0; k < 8; k++) lmax = fmaxf(lmax, tv[tt][k]);
    }

    // Row max: combine the two halves (same i), then across waves via LDS.
    lmax = fmaxf(lmax, __shfl_xor(lmax, 16, 32));
    if (lane < 16) atomicMax(&maxb[l15], fmono(lmax));
    __syncthreads();
    if (tid < 16) rmax_s[tid] = funmono(maxb[tid]);
    __syncthreads();

    // exp2(t - rowmax): bare v_exp_f32 per element; per-lane row partial sum.
    const float rmax = rmax_s[l15];
    float ps = 0.0f;
#pragma unroll
    for (int tt = 0; tt < 8; tt++) {
#pragma unroll
        for (int k = 0; k < 8; k++) {
            float e = __builtin_amdgcn_exp2f(tv[tt][k] - rmax);  // diag: exp2(-inf)=0
            tv[tt][k] = e;
            ps += e;
        }
    }
    ps += __shfl_xor(ps, 16, 32);
    if (lane < 16) atomicAdd(&rsum_s[l15], ps);   // ds_add_f32
    __syncthreads();
    if (tid < 16) inv_s[tid] = 1.0f / rsum_s[tid];
    __syncthreads();

    // Scaled write-out: 8 consecutive j per lane -> two b128 stores per tile.
    const float inv = inv_s[l15];
    float* rowp = out + (size_t)((b * Fn + f)) * NN * NN + (size_t)(i0 + l15) * NN;
#pragma unroll
    for (int tt = 0; tt < 8; tt++) {
        const int j0 = (wave * 8 + tt) * 16 + mbase;
        v4f q0, q1;
#pragma unroll
        for (int k = 0; k < 4; k++) { q0[k] = tv[tt][k] * inv; q1[k] = tv[tt][k + 4] * inv; }
        *(v4f*)(rowp + j0) = q0;
        *(v4f*)(rowp + j0 + 4) = q1;
    }
}

// ---------------------------------------------------------------------------
extern "C" void kernel_launch(void* const* d_in, const int* in_sizes, int n_in,
                              void* d_out, int out_size, void* d_ws, size_t ws_size,
                              hipStream_t stream) {
    (void)in_sizes; (void)n_in; (void)out_size; (void)ws_size;
    const float* emb   = (const float*)d_in[0];
    const float* alpha = (const float*)d_in[1];
    const float* beta  = (const float*)d_in[2];
    float* out = (float*)d_out;

    unsigned* dminb = (unsigned*)d_ws;                      // 32 slots
    float* pm = (float*)((char*)d_ws + 256);                // 32768 floats

    k_pm_init<<<dim3((Bn * Fn * NN + 255) / 256), dim3(256), 0, stream>>>(emb, alpha, pm, dminb);
    k_dmin<<<dim3(NN / 16, Bn), dim3(256), 0, stream>>>(emb, pm, dminb);
    k_softmax<<<dim3(NN / 16, Fn, Bn), dim3(256), 0, stream>>>(emb, beta, pm, dminb, out);
}